// MagnitudeAwarePooling_8297876816332
// MI455X (gfx1250) — compile-verified
//
#include <hip/hip_runtime.h>
#include <hip/hip_bf16.h>

typedef float v2f __attribute__((ext_vector_type(2)));
typedef float v8f __attribute__((ext_vector_type(8)));

#define B_    16
#define L_    4096
#define D_    1024
#define RPB   256          // L-rows per partial block
#define NTB   (L_ / RPB)   // 16 partial blocks per batch sample
#define TL    16           // tile rows (matches WMMA M/K granularity)
#define NT    (RPB / TL)   // 16 tiles per block
#define PITCH_F 1040       // padded LDS row pitch (floats): 16-bank row shift -> conflict-free column reads
#define TILE_F  (TL * PITCH_F)

// ws layout (floats)
#define ATTN_OFF 0
#define SUM_OFF  (256 * 1024)
#define MAX_OFF  (512 * 1024)
#define SCAL_OFF (768 * 1024)

// ---- CDNA5 async global->LDS copy (ASYNCcnt-tracked data path), inline asm for portability ----
__device__ __forceinline__ void async_ld16(unsigned lds_byte_off, const float* g) {
  asm volatile("global_load_async_to_lds_b128 %0, %1, off" :: "v"(lds_byte_off), "v"(g) : "memory");
}
__device__ __forceinline__ void wait_async_le16() {
  asm volatile("s_wait_asynccnt 0x10" ::: "memory");
}
__device__ __forceinline__ void wait_async_le0() {
  asm volatile("s_wait_asynccnt 0x0" ::: "memory");
}
__device__ __forceinline__ unsigned lds_off_of(const void* p) {
  // AMDGPU generic LDS address = (SHARED_BASE<<32) | byte-offset; low 32 bits are the DS offset
  return (unsigned)(unsigned long long)p;
}

// ============================ Kernel 1: fused streaming pass ============================
// grid = B_*NTB blocks, 256 threads (8 wave32). Each block: one b, 256 rows of L.
// Per 16-row tile (async-staged, double-buffered):
//   phase1: wave w computes full token energy of rows {2w,2w+1} (16x ds_b128 + 2 butterflies)
//           and accumulates max-pool partials for its column pattern.
//   phase2: lane r holds te[r]; butterfly max/sum -> online softmax state; p_r scalar per lane.
//   phase3: V_WMMA_F32_16X16X4_F32, A row0 = softmax numerators (via __shfl of p_r),
//           A row1 = ones (plain sum); C row M=0 rescaled online (M=8 row is zero -> safe).
__global__ __launch_bounds__(256) void map_pool_pass1(const float* __restrict__ x,
                                                      float* __restrict__ ws) {
  extern __shared__ float smem[];
  float* buf0   = smem;
  float* buf1   = smem + TILE_F;
  float* te_lds = smem + 2 * TILE_F;   // [TL] token energies of current tile

  const int tid = threadIdx.x;
  const int w   = tid >> 5;            // wave id 0..7 (wave32)
  const int ln  = tid & 31;            // lane id
  const int blk = blockIdx.x;          // 0..255
  const int b   = blk >> 4;            // / NTB
  const int pb  = blk & (NTB - 1);
  const size_t xbase = ((size_t)b * L_ + (size_t)pb * RPB) * D_;

  // prologue: async-stage tile 0 (each thread copies one 16B chunk per row)
  #pragma unroll
  for (int j = 0; j < TL; ++j) {
    async_ld16(lds_off_of(buf0 + j * PITCH_F + tid * 4),
               x + xbase + (size_t)j * D_ + tid * 4);
  }

  float runM = -3.0e38f, runZ = 0.0f, sqtot = 0.0f;
  float4 vmax[8];
  #pragma unroll
  for (int j = 0; j < 8; ++j) vmax[j] = make_float4(-3.0e38f, -3.0e38f, -3.0e38f, -3.0e38f);

  v8f c[8];
  const v8f vzero = {0.f, 0.f, 0.f, 0.f, 0.f, 0.f, 0.f, 0.f};
  #pragma unroll
  for (int i = 0; i < 8; ++i) c[i] = vzero;

  const int  m     = ln & 15;                  // WMMA M-row / B-col held by this lane
  const int  khalf = ln >> 4;                  // 0: K{0,1}  1: K{2,3}
  const bool am0   = (m == 0);
  const float abase = (m == 1) ? 1.0f : 0.0f;  // A row1 = ones, rows>=2 zero

  for (int t = 0; t < NT; ++t) {
    float* cur = (t & 1) ? buf1 : buf0;
    float* nxt = (t & 1) ? buf0 : buf1;

    if (t + 1 < NT) {
      const size_t nb = xbase + (size_t)(t + 1) * TL * D_;
      #pragma unroll
      for (int j = 0; j < TL; ++j) {
        async_ld16(lds_off_of(nxt + j * PITCH_F + tid * 4),
                   x + nb + (size_t)j * D_ + tid * 4);
      }
      wait_async_le16();   // tile t's 16 copies (issued first, done in-order) complete
    } else {
      wait_async_le0();
    }
    __syncthreads();

    // ---- phase 1: wave w owns rows 2w, 2w+1 fully; also max-pool partials ----
    #pragma unroll
    for (int rr = 0; rr < 2; ++rr) {
      const int r = 2 * w + rr;
      const float* rowp = cur + r * PITCH_F;
      float sq = 0.f;
      #pragma unroll
      for (int j = 0; j < 8; ++j) {
        const float4 v = *(const float4*)(rowp + j * 128 + ln * 4);
        sq += v.x * v.x + v.y * v.y + v.z * v.z + v.w * v.w;
        vmax[j].x = fmaxf(vmax[j].x, v.x);
        vmax[j].y = fmaxf(vmax[j].y, v.y);
        vmax[j].z = fmaxf(vmax[j].z, v.z);
        vmax[j].w = fmaxf(vmax[j].w, v.w);
      }
      sq += __shfl_xor(sq, 1, 32);
      sq += __shfl_xor(sq, 2, 32);
      sq += __shfl_xor(sq, 4, 32);
      sq += __shfl_xor(sq, 8, 32);
      sq += __shfl_xor(sq, 16, 32);
      if (ln == 0) te_lds[r] = sq;
    }
    __syncthreads();

    // ---- phase 2: online softmax state (lane r holds row r's energy) ----
    const float te_r = te_lds[m];   // rows duplicated across both lane halves
    float tmax = te_r;
    tmax = fmaxf(tmax, __shfl_xor(tmax, 1, 32));
    tmax = fmaxf(tmax, __shfl_xor(tmax, 2, 32));
    tmax = fmaxf(tmax, __shfl_xor(tmax, 4, 32));
    tmax = fmaxf(tmax, __shfl_xor(tmax, 8, 32));
    tmax = fmaxf(tmax, __shfl_xor(tmax, 16, 32));
    const float newM = fmaxf(runM, tmax);
    const float p_r  = __expf(te_r - newM);
    float psum = (ln < 16) ? p_r  : 0.f;
    float sqs  = (ln < 16) ? te_r : 0.f;
    psum += __shfl_xor(psum, 1, 32);  sqs += __shfl_xor(sqs, 1, 32);
    psum += __shfl_xor(psum, 2, 32);  sqs += __shfl_xor(sqs, 2, 32);
    psum += __shfl_xor(psum, 4, 32);  sqs += __shfl_xor(sqs, 4, 32);
    psum += __shfl_xor(psum, 8, 32);  sqs += __shfl_xor(sqs, 8, 32);
    psum += __shfl_xor(psum, 16, 32); sqs += __shfl_xor(sqs, 16, 32);
    const float scale = __expf(runM - newM);
    runZ  = runZ * scale + psum;
    runM  = newM;
    sqtot += sqs;

    // ---- phase 3: WMMA pools. C row M=0 = attn partial, row M=1 = plain sum. ----
    #pragma unroll
    for (int ct = 0; ct < 8; ++ct) c[ct][0] *= scale;   // rescale attn row (M=8 row is zero)

    const int ksel = khalf * 2;   // lane half -> K index pair {0,1} or {2,3}
    #pragma unroll
    for (int kc = 0; kc < 4; ++kc) {
      const int k0 = kc * 4;
      const float pa = __shfl(p_r, k0 + ksel, 32);       // p for K row in VGPR0
      const float pb = __shfl(p_r, k0 + ksel + 1, 32);   // p for K row in VGPR1
      const float a0 = am0 ? pa : abase;
      const float a1 = am0 ? pb : abase;
      const v2f a = {a0, a1};
      const float* colp = cur + (k0 + ksel) * PITCH_F + m;
      #pragma unroll
      for (int ct = 0; ct < 8; ++ct) {
        const int n0 = (w * 8 + ct) * 16;
        const float* bp = colp + n0;                     // B[k][n]: lane m = col n
        const v2f bv = {bp[0], bp[PITCH_F]};
        c[ct] = __builtin_amdgcn_wmma_f32_16x16x4_f32(
            false, a, false, bv, (short)0, c[ct], false, false);
      }
    }
    __syncthreads();   // protect 'cur' + te_lds before next tile overwrites
  }

  // ---- epilogue ----
  float* attw = ws + ATTN_OFF + (size_t)blk * 1024;
  float* sumw = ws + SUM_OFF  + (size_t)blk * 1024;
  float* maxw = ws + MAX_OFF  + (size_t)blk * 1024;
  if (ln < 16) {
    #pragma unroll
    for (int ct = 0; ct < 8; ++ct) {
      const int col = (w * 8 + ct) * 16 + ln;   // D row M=0 / M=1, col n = lane
      attw[col] = c[ct][0];
      sumw[col] = c[ct][1];
    }
  }
  // combine max-pool partials across waves via the (now free) tile buffer
  #pragma unroll
  for (int j = 0; j < 8; ++j) {
    *(float4*)(buf0 + w * 1024 + j * 128 + ln * 4) = vmax[j];
  }
  __syncthreads();
  {
    float4 mm = make_float4(-3.0e38f, -3.0e38f, -3.0e38f, -3.0e38f);
    #pragma unroll
    for (int ww = 0; ww < 8; ++ww) {
      const float4 v = *(const float4*)(buf0 + ww * 1024 + tid * 4);
      mm.x = fmaxf(mm.x, v.x);
      mm.y = fmaxf(mm.y, v.y);
      mm.z = fmaxf(mm.z, v.z);
      mm.w = fmaxf(mm.w, v.w);
    }
    *(float4*)(maxw + tid * 4) = mm;
  }
  if (tid == 0) {
    float* sc = ws + SCAL_OFF + blk * 4;
    sc[0] = runM; sc[1] = runZ; sc[2] = sqtot;
  }
}

// ============================ Kernel 2: combine + gate ============================
__global__ __launch_bounds__(256) void map_pool_pass2(const float* __restrict__ Wt,
                                                      const float* __restrict__ bias,
                                                      const float* __restrict__ ws,
                                                      float* __restrict__ out) {
  const int b   = blockIdx.x;
  const int tid = threadIdx.x;
  const float* sc = ws + SCAL_OFF + (size_t)(b * NTB) * 4;

  float M = -3.0e38f;
  #pragma unroll
  for (int i = 0; i < NTB; ++i) M = fmaxf(M, sc[i * 4 + 0]);
  float Z = 0.f, sq = 0.f;
  #pragma unroll
  for (int i = 0; i < NTB; ++i) {
    Z  += sc[i * 4 + 1] * __expf(sc[i * 4 + 0] - M);
    sq += sc[i * 4 + 2];
  }
  const float energy = sq * (1.0f / (float)(L_ * D_));
  const float lg0 = energy * Wt[0] + bias[0];
  const float lg1 = energy * Wt[1] + bias[1];
  const float lg2 = energy * Wt[2] + bias[2];
  const float mx  = fmaxf(lg0, fmaxf(lg1, lg2));
  const float e0 = __expf(lg0 - mx), e1 = __expf(lg1 - mx), e2 = __expf(lg2 - mx);
  const float inv = 1.0f / (e0 + e1 + e2);
  const float pw0 = e0 * inv, pw1 = e1 * inv, pw2 = e2 * inv;

  const int c0 = tid * 4;
  float4 aa = {0.f, 0.f, 0.f, 0.f};
  float4 ss = {0.f, 0.f, 0.f, 0.f};
  float4 mm = {-3.0e38f, -3.0e38f, -3.0e38f, -3.0e38f};
  for (int i = 0; i < NTB; ++i) {
    const int pblk = b * NTB + i;
    const float coef = __expf(sc[i * 4 + 0] - M);
    const float4 av = *(const float4*)(ws + ATTN_OFF + (size_t)pblk * 1024 + c0);
    const float4 sv = *(const float4*)(ws + SUM_OFF  + (size_t)pblk * 1024 + c0);
    const float4 mv = *(const float4*)(ws + MAX_OFF  + (size_t)pblk * 1024 + c0);
    aa.x += av.x * coef; aa.y += av.y * coef; aa.z += av.z * coef; aa.w += av.w * coef;
    ss.x += sv.x; ss.y += sv.y; ss.z += sv.z; ss.w += sv.w;
    mm.x = fmaxf(mm.x, mv.x); mm.y = fmaxf(mm.y, mv.y);
    mm.z = fmaxf(mm.z, mv.z); mm.w = fmaxf(mm.w, mv.w);
  }
  const float invZ = 1.0f / Z;
  const float invL = 1.0f / (float)L_;
  float4 o;
  o.x = pw0 * mm.x + pw1 * ss.x * invL + pw2 * aa.x * invZ;
  o.y = pw0 * mm.y + pw1 * ss.y * invL + pw2 * aa.y * invZ;
  o.z = pw0 * mm.z + pw1 * ss.z * invL + pw2 * aa.z * invZ;
  o.w = pw0 * mm.w + pw1 * ss.w * invL + pw2 * aa.w * invZ;
  *(float4*)(out + (size_t)b * D_ + c0) = o;
}

extern "C" void kernel_launch(void* const* d_in, const int* in_sizes, int n_in,
                              void* d_out, int out_size, void* d_ws, size_t ws_size,
                              hipStream_t stream) {
  const float* x    = (const float*)d_in[0];
  const float* W    = (const float*)d_in[1];   // [3,1]
  const float* bias = (const float*)d_in[2];   // [3]
  float* out = (float*)d_out;                  // [16,1024]
  float* ws  = (float*)d_ws;                   // ~3.15 MB used

  const size_t shmem = (size_t)(2 * TILE_F + TL) * sizeof(float);  // ~130 KB (<320 KB/WGP)
  map_pool_pass1<<<dim3(B_ * NTB), dim3(256), shmem, stream>>>(x, ws);
  map_pool_pass2<<<dim3(B_), dim3(256), 0, stream>>>(W, bias, ws, out);
}